// PathNN_21406117004232
// MI455X (gfx1250) — compile-verified
//
#include <hip/hip_runtime.h>
#include <math.h>

#define H 128
#define FOURH 512

typedef __attribute__((ext_vector_type(2))) float v2f;
typedef __attribute__((ext_vector_type(8))) float v8f;

__device__ __forceinline__ v8f wmma_f32(v2f a, v2f b, v8f c) {
    // D = A(16x4 f32) * B(4x16 f32) + C(16x16 f32)
    return __builtin_amdgcn_wmma_f32_16x16x4_f32(
        /*neg_a=*/false, a, /*neg_b=*/false, b,
        /*c_mod=*/(short)0, c, /*reuse_a=*/false, /*reuse_b=*/false);
}

__device__ __forceinline__ float sigmoidf_(float x) {
    return 1.0f / (1.0f + __expf(-x));
}

__device__ __forceinline__ float fast_tanhf_(float x) {
    float e = __expf(-2.0f * fabsf(x));
    float t = (1.0f - e) / (1.0f + e);
    return copysignf(t, x);
}

// Packed fragment-order layout for a K x Nw weight matrix:
//   element (K,N)  ->  ((K>>2)*Nw + N)*4 + (K&3)
// so a lane's B-fragment pair (K=k+kh*2, K=k+kh*2+1) at column n is one
// contiguous float2 at  ((k>>2)*Nw + n)*4 + kh*2   (8-byte aligned).
__device__ __forceinline__ v2f load_bfrag(const float* __restrict__ Wp, int Nw,
                                          int k, int kh, int n) {
    return *(const v2f*)(Wp + ((size_t)((k >> 2) * Nw + n) << 2) + kh * 2);
}

// ---------------------------------------------------------------------------
// Prep: pack all weight matrices into WMMA fragment order, bias = b_ih + b_hh.
//   W_ih/W_hh given as [512,128] (gates x input)  -> B operand is W^T [128,512]
//   W1/W2 given as [128,128] (input x output)     -> B operand directly
// ---------------------------------------------------------------------------
__global__ void prep_kernel(const float* __restrict__ Wih, const float* __restrict__ Whh,
                            const float* __restrict__ W1,  const float* __restrict__ W2,
                            const float* __restrict__ bih, const float* __restrict__ bhh,
                            float* __restrict__ WihP, float* __restrict__ WhhP,
                            float* __restrict__ W1P,  float* __restrict__ W2P,
                            float* __restrict__ biasc) {
    int idx = blockIdx.x * blockDim.x + threadIdx.x;
    if (idx < FOURH * H) {
        int n = idx / H;    // gate-output index 0..511  (= N of B operand)
        int k = idx % H;    // input index 0..127        (= K of B operand)
        int dst = (((k >> 2) * FOURH + n) << 2) + (k & 3);
        WihP[dst] = Wih[idx];
        WhhP[dst] = Whh[idx];
    }
    if (idx < H * H) {
        int k = idx / H;    // W1[k][n]: K x N already
        int n = idx % H;
        int dst = (((k >> 2) * H + n) << 2) + (k & 3);
        W1P[dst] = W1[idx];
        W2P[dst] = W2[idx];
    }
    if (idx < FOURH) biasc[idx] = bih[idx] + bhh[idx];
}

// ---------------------------------------------------------------------------
// Encoder GEMM: out[r,:] = relu(A[r,:] @ W + b), A:[nrows,128], Wp packed 128x128
// 32 rows per block, 8 waves: wave -> (mt = w>>2) x (2 N-tiles at (w&3)*32)
// ---------------------------------------------------------------------------
__global__ __launch_bounds__(256)
void encoder_kernel(const float* __restrict__ A, const float* __restrict__ Wp,
                    const float* __restrict__ bias, float* __restrict__ out, int nrows) {
    __shared__ float s_a[32][132];   // pad to 132 to avoid LDS bank conflicts

    int tid  = threadIdx.x;
    int row0 = blockIdx.x * 32;

    #pragma unroll
    for (int it = 0; it < 4; ++it) {
        int lin = it * 256 + tid;            // 0..1023
        int r   = lin >> 5;                  // 0..31
        int seg = lin & 31;                  // float4 segment
        int gr  = row0 + r; if (gr >= nrows) gr = nrows - 1;
        float4 v = ((const float4*)(A + (size_t)gr * H))[seg];
        s_a[r][seg * 4 + 0] = v.x; s_a[r][seg * 4 + 1] = v.y;
        s_a[r][seg * 4 + 2] = v.z; s_a[r][seg * 4 + 3] = v.w;
    }
    __syncthreads();

    int wave = tid >> 5, lane = tid & 31;
    int mt = wave >> 2;            // M sub-tile (rows mt*16..mt*16+15)
    int nb = (wave & 3) * 32;      // base output column
    int lm = lane & 15;
    int kh = lane >> 4;            // K-half select (0 or 1)

    v8f acc[2] = {};
    for (int k = 0; k < H; k += 4) {
        v2f a = *(const v2f*)(&s_a[mt * 16 + lm][k + kh * 2]);
        #pragma unroll
        for (int nt = 0; nt < 2; ++nt) {
            v2f b = load_bfrag(Wp, H, k, kh, nb + nt * 16 + lm);
            acc[nt] = wmma_f32(a, b, acc[nt]);
        }
    }

    #pragma unroll
    for (int nt = 0; nt < 2; ++nt) {
        int col = nb + nt * 16 + lm;
        float bv = bias[col];
        #pragma unroll
        for (int g = 0; g < 8; ++g) {
            int r  = mt * 16 + g + kh * 8;
            int gr = row0 + r;
            if (gr < nrows) {
                float v = acc[nt][g] + bv;
                out[(size_t)gr * H + col] = fmaxf(v, 0.0f);
            }
        }
    }
}

// ---------------------------------------------------------------------------
// Fused PathConv LSTM: 32 paths per block, all L steps in LDS, then scatter-add.
// gates[TP=32, 512] split over 8 waves (64 cols each; 64 | 128 so each wave's
// columns live inside a single gate i/f/g/o).
// ---------------------------------------------------------------------------
template<int L>
__global__ __launch_bounds__(256)
void path_lstm_kernel(const float* __restrict__ h_in, const int* __restrict__ paths,
                      const float* __restrict__ WihP, const float* __restrict__ WhhP,
                      const float* __restrict__ biasc, float* __restrict__ agg, int P) {
    __shared__ float s_x[32][132];
    __shared__ float s_h[32][132];
    __shared__ float s_c[32][132];
    __shared__ float s_g[32][520];
    __shared__ int   s_idx[32];

    int tid  = threadIdx.x;
    int p0   = blockIdx.x * 32;
    int wave = tid >> 5, lane = tid & 31;
    int nb = wave * 64;            // this wave's gate-column base (0..448)
    int lm = lane & 15;
    int kh = lane >> 4;

    #pragma unroll
    for (int it = 0; it < 16; ++it) {
        int lin = it * 256 + tid;
        int r = lin >> 7, j = lin & 127;
        s_c[r][j] = 0.0f;
        s_h[r][j] = 0.0f;
    }

    for (int t = 0; t < L; ++t) {
        __syncthreads();                       // s_h/s_c updates + s_x reuse
        if (tid < 32) {
            int pp = p0 + tid; if (pp >= P) pp = P - 1;
            s_idx[tid] = paths[(size_t)pp * L + t];
        }
        __syncthreads();

        // gather x_t = h_in[paths[:,t]]
        #pragma unroll
        for (int it = 0; it < 4; ++it) {
            int lin = it * 256 + tid;
            int r = lin >> 5, seg = lin & 31;
            int src = s_idx[r];
            float4 v = ((const float4*)(h_in + (size_t)src * H))[seg];
            s_x[r][seg * 4 + 0] = v.x; s_x[r][seg * 4 + 1] = v.y;
            s_x[r][seg * 4 + 2] = v.z; s_x[r][seg * 4 + 3] = v.w;
        }
        __syncthreads();

        v8f acc[2][4] = {};
        // pass 1: x_t @ Wih^T
        for (int k = 0; k < H; k += 4) {
            v2f a0 = *(const v2f*)(&s_x[lm][k + kh * 2]);
            v2f a1 = *(const v2f*)(&s_x[16 + lm][k + kh * 2]);
            #pragma unroll
            for (int nt = 0; nt < 4; ++nt) {
                v2f b = load_bfrag(WihP, FOURH, k, kh, nb + nt * 16 + lm);
                acc[0][nt] = wmma_f32(a0, b, acc[0][nt]);
                acc[1][nt] = wmma_f32(a1, b, acc[1][nt]);
            }
        }
        // pass 2: h_{t-1} @ Whh^T (h == 0 at t==0)
        if (t > 0) {
            for (int k = 0; k < H; k += 4) {
                v2f a0 = *(const v2f*)(&s_h[lm][k + kh * 2]);
                v2f a1 = *(const v2f*)(&s_h[16 + lm][k + kh * 2]);
                #pragma unroll
                for (int nt = 0; nt < 4; ++nt) {
                    v2f b = load_bfrag(WhhP, FOURH, k, kh, nb + nt * 16 + lm);
                    acc[0][nt] = wmma_f32(a0, b, acc[0][nt]);
                    acc[1][nt] = wmma_f32(a1, b, acc[1][nt]);
                }
            }
        }
        // gates (+ combined bias) -> LDS
        #pragma unroll
        for (int mtl = 0; mtl < 2; ++mtl) {
            #pragma unroll
            for (int nt = 0; nt < 4; ++nt) {
                int col = nb + nt * 16 + lm;
                float bv = biasc[col];
                #pragma unroll
                for (int g = 0; g < 8; ++g) {
                    int r = mtl * 16 + g + kh * 8;
                    s_g[r][col] = acc[mtl][nt][g] + bv;
                }
            }
        }
        __syncthreads();

        // elementwise LSTM cell update (gate order i,f,g,o)
        #pragma unroll
        for (int it = 0; it < 16; ++it) {
            int lin = it * 256 + tid;
            int r = lin >> 7, j = lin & 127;
            float gi = s_g[r][j];
            float gf = s_g[r][H + j];
            float gg = s_g[r][2 * H + j];
            float go = s_g[r][3 * H + j];
            float i_ = sigmoidf_(gi);
            float f_ = sigmoidf_(gf);
            float g_ = fast_tanhf_(gg);
            float o_ = sigmoidf_(go);
            float c  = f_ * s_c[r][j] + i_ * g_;
            s_c[r][j] = c;
            s_h[r][j] = o_ * fast_tanhf_(c);
        }
    }
    __syncthreads();

    // scatter-add h_last into agg at paths[:, L-1] (s_idx holds t = L-1)
    #pragma unroll
    for (int it = 0; it < 16; ++it) {
        int lin = it * 256 + tid;
        int r = lin >> 7, j = lin & 127;
        if (p0 + r < P) {
            int dest = s_idx[r];
            unsafeAtomicAdd(&agg[(size_t)dest * H + j], s_h[r][j]);
        }
    }
}

// ---------------------------------------------------------------------------
// out = relu(agg + h)   (elementwise; safe in place for out==agg)
// ---------------------------------------------------------------------------
__global__ void residual_relu_kernel(const float* __restrict__ agg,
                                     const float* __restrict__ hres,
                                     float* __restrict__ out, int n) {
    int i = blockIdx.x * blockDim.x + threadIdx.x;
    if (i < n) out[i] = fmaxf(agg[i] + hres[i], 0.0f);
}

// ---------------------------------------------------------------------------
extern "C" void kernel_launch(void* const* d_in, const int* in_sizes, int n_in,
                              void* d_out, int out_size, void* d_ws, size_t ws_size,
                              hipStream_t stream) {
    const float* x     = (const float*)d_in[0];
    const int*   path2 = (const int*)  d_in[1];
    const int*   path3 = (const int*)  d_in[2];
    const float* W1    = (const float*)d_in[3];
    const float* b1    = (const float*)d_in[4];
    const float* W2    = (const float*)d_in[5];
    const float* b2    = (const float*)d_in[6];
    const float* Wih   = (const float*)d_in[7];
    const float* Whh   = (const float*)d_in[8];
    const float* bih   = (const float*)d_in[9];
    const float* bhh   = (const float*)d_in[10];
    float* out = (float*)d_out;

    const int N  = in_sizes[0] / H;     // 50000
    const int P2 = in_sizes[1] / 2;     // 200000
    const int P3 = in_sizes[2] / 3;     // 200000

    float* w     = (float*)d_ws;
    float* WihP  = w;                         // 128*512
    float* WhhP  = WihP + FOURH * H;          // 128*512
    float* W1P   = WhhP + FOURH * H;          // 128*128
    float* W2P   = W1P + H * H;               // 128*128
    float* biasc = W2P + H * H;               // 512
    float* bufA  = biasc + FOURH;             // N*128  (16B-aligned: offset 164352 floats)
    float* bufB  = bufA + (size_t)N * H;      // N*128

    prep_kernel<<<(FOURH * H + 255) / 256, 256, 0, stream>>>(
        Wih, Whh, W1, W2, bih, bhh, WihP, WhhP, W1P, W2P, biasc);

    int encBlocks = (N + 31) / 32;
    // h0 = relu(x@W1+b1) -> bufA ; h_enc = relu(h0@W2+b2) -> bufB
    encoder_kernel<<<encBlocks, 256, 0, stream>>>(x,    W1P, b1, bufA, N);
    encoder_kernel<<<encBlocks, 256, 0, stream>>>(bufA, W2P, b2, bufB, N);

    // layer 1: agg in bufA, h in bufB; result (in place) -> bufA
    hipMemsetAsync(bufA, 0, (size_t)N * H * sizeof(float), stream);
    path_lstm_kernel<2><<<(P2 + 31) / 32, 256, 0, stream>>>(bufB, path2, WihP, WhhP,
                                                            biasc, bufA, P2);
    residual_relu_kernel<<<(N * H + 255) / 256, 256, 0, stream>>>(bufA, bufB, bufA, N * H);

    // layer 2: agg in bufB, h in bufA; result -> d_out
    hipMemsetAsync(bufB, 0, (size_t)N * H * sizeof(float), stream);
    path_lstm_kernel<3><<<(P3 + 31) / 32, 256, 0, stream>>>(bufA, path3, WihP, WhhP,
                                                            biasc, bufB, P3);
    residual_relu_kernel<<<(N * H + 255) / 256, 256, 0, stream>>>(bufB, bufA, out, N * H);
}